// LowFreqPenaltyLoss_80290118631926
// MI455X (gfx1250) — compile-verified
//
#include <hip/hip_runtime.h>

// ---------------------------------------------------------------------------
// LowFreqPenaltyLoss on MI455X (gfx1250)
//
// Roofline: input = 256*3*256*256 fp32 = 201.3 MB -> 8.6 us at 23.3 TB/s.
// Factored DCT contraction = ~0.83 GFLOP -> ~96 TFLOP/s needed; trivially
// covered by v_wmma_f32_16x16x4_f32, so the kernel is designed around
// streaming HBM once: TDM (tensor_load_to_lds) stages 16-row slabs into a
// double-buffered LDS region, WMMA consumes fragments from LDS.
// ---------------------------------------------------------------------------

typedef float v2f __attribute__((ext_vector_type(2)));
typedef float v8f __attribute__((ext_vector_type(8)));
typedef unsigned int u32x4 __attribute__((ext_vector_type(4)));
typedef int i32x4 __attribute__((ext_vector_type(4)));
typedef int i32x8 __attribute__((ext_vector_type(8)));

#define IMG_H 256
#define IMG_W 256
#define LOWF 8
#define SLAB_ROWS 16
#define NSLABS (IMG_H / SLAB_ROWS)
#define PI_F 3.14159265358979323846f

// Issue one TDM descriptor: copy `nrows` x 256 fp32 rows (row stride 256)
// from global `gsrc` into LDS at byte offset `lds_byte_off`.
// D# group0: count=1, lds_addr[63:32], global_addr[120:64], type=2[127:126].
// D# group1: data_size=4B, tensor_dim0=256, tensor_dim1=nrows,
//            tile_dim0=256, tile_dim1=nrows, tensor_dim0_stride=256.
__device__ __forceinline__ void tdm_load_rows(const float* gsrc,
                                              unsigned lds_byte_off,
                                              int nrows) {
  unsigned long long ga = (unsigned long long)(uintptr_t)gsrc;
  u32x4 g0;
  g0[0] = 1u;                                    // count=1, user mode
  g0[1] = lds_byte_off;                          // lds_addr (bytes)
  g0[2] = (unsigned)(ga & 0xFFFFFFFFu);          // global_addr[31:0]
  g0[3] = (unsigned)((ga >> 32) & 0x1FFFFFFu)    // global_addr[56:32]
          | (2u << 30);                          // type=2 ("image")
  i32x8 g1;
  g1[0] = 0x00020000;                            // data_size=2 -> 4 bytes
  g1[1] = (int)(256u << 16);                     // tensor_dim0=256 (bits 63:48)
  g1[2] = (int)((unsigned)nrows << 16);          // tensor_dim1 (bits 95:80)
  g1[3] = (int)(256u << 16);                     // tile_dim0=256 (bits 127:112)
  g1[4] = nrows;                                 // tile_dim1 (bits 143:128)
  g1[5] = 256;                                   // tensor_dim0_stride (bits 191:160)
  g1[6] = 0;
  g1[7] = 0;
  i32x4 z4 = {0, 0, 0, 0};                       // groups 2/3 unused (2D tile)
  i32x8 z8 = {0, 0, 0, 0, 0, 0, 0, 0};
  __builtin_amdgcn_tensor_load_to_lds(g0, g1, z4, z4, z8, 0);
}

__global__ void __launch_bounds__(128)
lowfreq_main_kernel(const float* __restrict__ delta, float* __restrict__ out,
                    float scale) {
  // LDS: DCT table [16][256] (rows 8..15 zero) + double slab buffer
  // [2][16][256] (reused later as T storage [16 strips][16][16]) + reducer.
  __shared__ float smem[4096 + 8192 + 32];
  float* sC = smem;            // 16 KB
  float* sBuf = smem + 4096;   // 32 KB
  float* sRed = smem + 4096 + 8192;

  const int tid = threadIdx.x;
  const int lane = tid & 31;
  const int laneLo = lane & 15;
  const int laneHi = lane >> 4;                       // 0 or 1
  const int wave = __builtin_amdgcn_readfirstlane(tid >> 5);  // 0..3

  // Unnormalized DCT-II basis: C[a][h] = 2*cos(pi*(2h+1)*a/(2*256)).
  // Rows a>=8 zero-padded so the 16x16 WMMA padding produces exact zeros.
  for (int idx = tid; idx < 16 * 256; idx += 128) {
    int a = idx >> 8, h = idx & 255;
    float v = 0.0f;
    if (a < LOWF)
      v = 2.0f * __cosf(PI_F * (2.0f * (float)h + 1.0f) * (float)a * (1.0f / 512.0f));
    sC[idx] = v;
  }
  if (tid == 0) *sRed = 0.0f;

  const float* img = delta + (size_t)blockIdx.x * (IMG_H * IMG_W);

  // Prologue: each wave DMAs its own 4 rows of slab 0 (quarter-slab per wave
  // keeps TENSORcnt waits wave-local; __syncthreads publishes to the block).
  tdm_load_rows(img + (size_t)(wave * 4) * IMG_W,
                (unsigned)(uintptr_t)(sBuf + wave * 4 * IMG_W), 4);

  // Stage 1: T_strip[16x16] = sum_h Ch[16x4] * D[4x16], per wave 4 strips.
  v8f acc[4];
#pragma unroll
  for (int j = 0; j < 4; ++j) acc[j] = (v8f){0.f, 0.f, 0.f, 0.f, 0.f, 0.f, 0.f, 0.f};

  for (int slab = 0; slab < NSLABS; ++slab) {
    const int buf = slab & 1;
    if (slab + 1 < NSLABS) {
      tdm_load_rows(img + (size_t)((slab + 1) * SLAB_ROWS + wave * 4) * IMG_W,
                    (unsigned)(uintptr_t)(sBuf + (1 - buf) * 4096 + wave * 4 * IMG_W), 4);
      __builtin_amdgcn_s_wait_tensorcnt(1);  // slab `slab` done, slab+1 in flight
    } else {
      __builtin_amdgcn_s_wait_tensorcnt(0);
    }
    __syncthreads();  // all quarter-slabs (and, on iter 0, the table) visible

    const float* bbase = sBuf + buf * 4096;
#pragma unroll
    for (int kl = 0; kl < SLAB_ROWS; kl += 4) {
      const int h0 = slab * SLAB_ROWS + kl;
      // A frag (Ch 16x4): lanes0-15 hold K={0,1}, lanes16-31 K={2,3}.
      v2f afrag = *(const v2f*)(sC + laneLo * 256 + h0 + 2 * laneHi);
#pragma unroll
      for (int j = 0; j < 4; ++j) {
        const int w0 = (wave * 4 + j) * 16;
        // B frag (D 4x16): v0 = D[k0+2*laneHi][w0+laneLo], v1 = next row.
        const float* bp = bbase + (kl + 2 * laneHi) * IMG_W + w0 + laneLo;
        v2f bfrag = {bp[0], bp[IMG_W]};
        acc[j] = __builtin_amdgcn_wmma_f32_16x16x4_f32(
            false, afrag, false, bfrag, (short)0, acc[j], false, false);
      }
    }
    __syncthreads();  // done reading this buffer before it is DMA-overwritten
  }

  // Spill T strips (C/D layout -> row-major [16][16] in LDS, slab bufs dead).
#pragma unroll
  for (int j = 0; j < 4; ++j) {
    const int sidx = wave * 4 + j;
    float* tb = sBuf + sidx * 256;
#pragma unroll
    for (int r = 0; r < 8; ++r)
      tb[(r + 8 * laneHi) * 16 + laneLo] = acc[j][r];
  }
  __syncthreads();

  // Stage 2: O[16x16] += T_chunk[16x4] * CwT_chunk[4x16] over all 16 w's of
  // each strip; each wave accumulates its 4 strips, block-combined via LDS.
  v8f oacc = (v8f){0.f, 0.f, 0.f, 0.f, 0.f, 0.f, 0.f, 0.f};
#pragma unroll
  for (int j = 0; j < 4; ++j) {
    const int sidx = wave * 4 + j;
    const int w0 = sidx * 16;
    const float* tb = sBuf + sidx * 256;
#pragma unroll
    for (int k4 = 0; k4 < 16; k4 += 4) {
      v2f afrag = *(const v2f*)(tb + laneLo * 16 + k4 + 2 * laneHi);
      v2f bfrag = *(const v2f*)(sC + laneLo * 256 + w0 + k4 + 2 * laneHi);
      oacc = __builtin_amdgcn_wmma_f32_16x16x4_f32(
          false, afrag, false, bfrag, (short)0, oacc, false, false);
    }
  }

  // |O| sum: padded rows/cols are exact zeros, so summing all 16x16 is fine.
  float s = 0.0f;
#pragma unroll
  for (int r = 0; r < 8; ++r) s += __builtin_fabsf(oacc[r]);
#pragma unroll
  for (int off = 16; off >= 1; off >>= 1) s += __shfl_xor(s, off, 32);
  if (lane == 0) atomicAdd(sRed, s);
  __syncthreads();
  if (tid == 0) atomicAdd(out, (*sRed) * scale);
}

__global__ void lowfreq_zero_kernel(float* out) {
  if (threadIdx.x == 0 && blockIdx.x == 0) out[0] = 0.0f;
}

extern "C" void kernel_launch(void* const* d_in, const int* in_sizes, int n_in,
                              void* d_out, int out_size, void* d_ws, size_t ws_size,
                              hipStream_t stream) {
  (void)n_in; (void)out_size; (void)d_ws; (void)ws_size;
  const float* delta = (const float*)d_in[0];
  float* out = (float*)d_out;
  const int images = in_sizes[0] / (IMG_H * IMG_W);  // B*C = 768
  const float scale = 1.0f / ((float)images * (float)(LOWF * LOWF));
  lowfreq_zero_kernel<<<1, 32, 0, stream>>>(out);
  lowfreq_main_kernel<<<images, 128, 0, stream>>>(delta, out, scale);
}